// PixelContrastLoss_22436909154959
// MI455X (gfx1250) — compile-verified
//
#include <hip/hip_runtime.h>
#include <math.h>

// ---------------------------------------------------------------------------
// Supervised-contrastive loss on MI455X (gfx1250), fp32 via V_WMMA_F32_16X16X4_F32
// X treated as F[8192][256] (class-major; loss is row-permutation invariant),
// label of row r is y[r >> 6].
// K2: 8 waves/WG own 8 row-blocks and SHARE column tiles through double-
//     buffered LDS (8x less L2 traffic, LDS rows padded to kill bank conflicts).
// ---------------------------------------------------------------------------

#define C_CLASSES 128
#define V_VIEWS   64
#define D_DIM     256
#define N_ROWS    (C_CLASSES * V_VIEWS)   // 8192
#define N_TILES   (N_ROWS / 16)           // 512 column tiles of 16
#define KF        (D_DIM / 4)             // 64 WMMA k-steps (K=4 each)
#define BPAD      260                     // padded LDS row (bank-conflict free)
#define N_CHUNK   8                       // J chunks (one per WG group)
#define TILES_PER_CHUNK (N_TILES / N_CHUNK)  // 64
#define INV_T     10.0f                   // 1 / 0.1
#define T_OVER_B  (0.1f / 0.07f)

typedef float v2f __attribute__((ext_vector_type(2)));
typedef float v8f __attribute__((ext_vector_type(8)));

__device__ __forceinline__ v8f wmma4(v2f a, v2f b, v8f c) {
  // D = A(16x4,f32) * B(4x16,f32) + C(16x16,f32)
  return __builtin_amdgcn_wmma_f32_16x16x4_f32(
      /*neg_a=*/false, a, /*neg_b=*/false, b,
      /*c_mod=*/(short)0, c, /*reuse_a=*/false, /*reuse_b=*/false);
}

// --------------------------- K1: diagonal shifts ---------------------------
__global__ __launch_bounds__(256) void k_diag(const float* __restrict__ X,
                                              float* __restrict__ s) {
  int i = blockIdx.x * blockDim.x + threadIdx.x;
  if (i >= N_ROWS) return;
  const float4* p = (const float4*)(X + (size_t)i * D_DIM);
  float acc = 0.0f;
#pragma unroll
  for (int k = 0; k < D_DIM / 4; ++k) {
    float4 v = p[k];
    acc = fmaf(v.x, v.x, acc);
    acc = fmaf(v.y, v.y, acc);
    acc = fmaf(v.z, v.z, acc);
    acc = fmaf(v.w, v.w, acc);
  }
  s[i] = acc * INV_T;  // shift in logit space (substitutes for row max)
}

// ----------------- K2: partial exp-sums with LDS-shared B tiles -------------
// Grid: 512 WGs = 64 row-groups x 8 J-chunks. Wave w owns row block
// (rg*8 + w); all 8 waves consume the SAME column tiles, staged once per WG
// into double-buffered LDS. A-block stays register-resident per wave.
__global__ __launch_bounds__(256) void k_negsum(const float* __restrict__ X,
                                                const int* __restrict__ y,
                                                const float* __restrict__ s,
                                                float* __restrict__ negpart) {
  __shared__ float bbuf[2][16][BPAD];
  __shared__ float ldsT[8][32][8];
  __shared__ float ldsP[8][32][8];

  const int tid  = threadIdx.x;
  const int wave = tid >> 5;
  const int lane = tid & 31;
  const int half = lane >> 4;
  const int l16  = lane & 15;

  const int rg      = blockIdx.x >> 3;      // row group 0..63
  const int chunk   = blockIdx.x & 7;       // J chunk 0..7
  const int rb      = rg * 8 + wave;        // this wave's 16-row block
  const int rowBase = rb * 16;
  const int myLabel = y[rb >> 2];           // 4 row-blocks per class
  const int jt0     = chunk * TILES_PER_CHUNK;

  // Register-resident A fragments: row (rowBase + l16), k pair by lane half
  const float* arow = X + (size_t)(rowBase + l16) * D_DIM + 2 * half;
  v2f aF[KF];
#pragma unroll
  for (int kf = 0; kf < KF; ++kf) aF[kf] = *(const v2f*)(arow + kf * 4);

  float sh[8];
#pragma unroll
  for (int e = 0; e < 8; ++e) sh[e] = s[rowBase + e + 8 * half];

  float tot[8] = {0, 0, 0, 0, 0, 0, 0, 0};
  float pos[8] = {0, 0, 0, 0, 0, 0, 0, 0};

  // staging map: this thread copies 4 float4s of the 16x256 tile
  const int srow0 = tid >> 6;            // base row for q=0 (rows advance by 4)
  const int scol  = (tid & 63) * 4;      // float offset within row

  // prime: stage tile jt0 into buffer 0
  float4 pre[4];
#pragma unroll
  for (int q = 0; q < 4; ++q)
    pre[q] = *(const float4*)(X + (size_t)(jt0 * 16 + srow0 + 4 * q) * D_DIM + scol);
#pragma unroll
  for (int q = 0; q < 4; ++q)
    *(float4*)&bbuf[0][srow0 + 4 * q][scol] = pre[q];
  __syncthreads();

  for (int i = 0; i < TILES_PER_CHUNK; ++i) {
    const int jt = jt0 + i;
    // issue global prefetch of next tile (consumed after compute)
    if (i + 1 < TILES_PER_CHUNK) {
#pragma unroll
      for (int q = 0; q < 4; ++q)
        pre[q] = *(const float4*)(X + (size_t)((jt + 1) * 16 + srow0 + 4 * q) * D_DIM + scol);
    }

    // compute this wave's 16x16 tile from LDS-staged B
    const float* bB = &bbuf[i & 1][l16][2 * half];
    v8f c0 = {0, 0, 0, 0, 0, 0, 0, 0};
    v8f c1 = {0, 0, 0, 0, 0, 0, 0, 0};
#pragma unroll
    for (int kf = 0; kf < KF; kf += 2) {
      v2f b0 = *(const v2f*)(bB + kf * 4);
      v2f b1 = *(const v2f*)(bB + kf * 4 + 4);
      c0 = wmma4(aF[kf],     b0, c0);
      c1 = wmma4(aF[kf + 1], b1, c1);
    }
    const bool isPos = (y[jt >> 2] == myLabel);  // whole tile is one class
#pragma unroll
    for (int e = 0; e < 8; ++e) {
      float l  = (c0[e] + c1[e]) * INV_T - sh[e];
      float ex = __expf(l);
      tot[e] += ex;
      if (isPos) pos[e] += ex;
    }

    // drain prefetch into the other buffer (loads have had compute to hide)
    if (i + 1 < TILES_PER_CHUNK) {
#pragma unroll
      for (int q = 0; q < 4; ++q)
        *(float4*)&bbuf[(i + 1) & 1][srow0 + 4 * q][scol] = pre[q];
    }
    __syncthreads();
  }

  // per-wave cross-lane reduction (each wave owns its own 16 rows)
#pragma unroll
  for (int e = 0; e < 8; ++e) {
    ldsT[wave][lane][e] = tot[e];
    ldsP[wave][lane][e] = pos[e];
  }
  __syncthreads();

  if (lane < 16) {
    const int m     = lane;
    const int e     = m & 7;
    const int lbase = (m >> 3) * 16;
    float t = 0.0f, p = 0.0f;
    for (int l = 0; l < 16; ++l) {
      t += ldsT[wave][lbase + l][e];
      p += ldsP[wave][lbase + l][e];
    }
    negpart[(size_t)chunk * N_ROWS + rowBase + m] = t - p;  // exclude positives
  }
}

// ---------------- K2b: reduce the 8 J-chunk partials (deterministic) --------
__global__ __launch_bounds__(256) void k_negred(const float* __restrict__ negpart,
                                                float* __restrict__ negsum) {
  int i = blockIdx.x * blockDim.x + threadIdx.x;
  if (i >= N_ROWS) return;
  float a = 0.0f;
#pragma unroll
  for (int c = 0; c < N_CHUNK; ++c) a += negpart[(size_t)c * N_ROWS + i];
  negsum[i] = a;
}

// ------- K3: positives pass: sum over pos j of l - log(exp(l)+negsum) -------
__global__ __launch_bounds__(256) void k_pos(const float* __restrict__ X,
                                             const int* __restrict__ y,
                                             const float* __restrict__ s,
                                             const float* __restrict__ negsum,
                                             float* __restrict__ partial) {
  __shared__ float ldsAcc[8][32][8];
  __shared__ float rowRed[16];

  const int tid  = threadIdx.x;
  const int wave = tid >> 5;
  const int lane = tid & 31;
  const int half = lane >> 4;
  const int l16  = lane & 15;

  const int rb      = blockIdx.x;
  const int rowBase = rb * 16;
  const int myLabel = y[rb >> 2];

  // positive count per row: V * (#classes with this label)
  int kcnt = 0;
  for (int cc = 0; cc < C_CLASSES; ++cc) kcnt += (y[cc] == myLabel) ? 1 : 0;
  const float invCnt = 1.0f / (float)(kcnt * V_VIEWS);

  const float* arow = X + (size_t)(rowBase + l16) * D_DIM + 2 * half;
  v2f aF[KF];
#pragma unroll
  for (int kf = 0; kf < KF; ++kf) aF[kf] = *(const v2f*)(arow + kf * 4);

  float sh[8], ns[8];
#pragma unroll
  for (int e = 0; e < 8; ++e) {
    sh[e] = s[rowBase + e + 8 * half];
    ns[e] = negsum[rowBase + e + 8 * half];
  }

  float acc[8] = {0, 0, 0, 0, 0, 0, 0, 0};

  for (int jt = wave; jt < N_TILES; jt += 8) {
    if (y[jt >> 2] != myLabel) continue;  // only positive tiles (~5%)
    const float* brow = X + (size_t)(jt * 16 + l16) * D_DIM + 2 * half;
    v8f c0 = {0, 0, 0, 0, 0, 0, 0, 0};
    v8f c1 = {0, 0, 0, 0, 0, 0, 0, 0};
#pragma unroll
    for (int kf = 0; kf < KF; kf += 2) {
      v2f b0 = *(const v2f*)(brow + kf * 4);
      v2f b1 = *(const v2f*)(brow + kf * 4 + 4);
      c0 = wmma4(aF[kf],     b0, c0);
      c1 = wmma4(aF[kf + 1], b1, c1);
    }
#pragma unroll
    for (int e = 0; e < 8; ++e) {
      float l = (c0[e] + c1[e]) * INV_T - sh[e];
      acc[e] += l - __logf(__expf(l) + ns[e]);
    }
  }

#pragma unroll
  for (int e = 0; e < 8; ++e) ldsAcc[wave][lane][e] = acc[e];
  __syncthreads();

  if (tid < 16) {
    const int m     = tid;
    const int e     = m & 7;
    const int lbase = (m >> 3) * 16;
    float a = 0.0f;
    for (int w = 0; w < 8; ++w)
      for (int l = 0; l < 16; ++l) a += ldsAcc[w][lbase + l][e];
    rowRed[m] = a * invCnt;  // mean_log_prob_pos for this row
  }
  __syncthreads();

  if (tid == 0) {
    float sum = 0.0f;
    for (int m = 0; m < 16; ++m) sum += rowRed[m];
    partial[rb] = sum;
  }
}

// --------------------------- K4: final reduce ------------------------------
__global__ __launch_bounds__(256) void k_final(const float* __restrict__ partial,
                                               float* __restrict__ out) {
  __shared__ float red[256];
  float a = 0.0f;
  for (int i = threadIdx.x; i < N_TILES; i += 256) a += partial[i];
  red[threadIdx.x] = a;
  __syncthreads();
  for (int st = 128; st > 0; st >>= 1) {
    if (threadIdx.x < st) red[threadIdx.x] += red[threadIdx.x + st];
    __syncthreads();
  }
  if (threadIdx.x == 0)
    out[0] = -T_OVER_B * red[0] / (float)N_ROWS;
}

// ---------------------------------------------------------------------------
extern "C" void kernel_launch(void* const* d_in, const int* in_sizes, int n_in,
                              void* d_out, int out_size, void* d_ws, size_t ws_size,
                              hipStream_t stream) {
  (void)in_sizes; (void)n_in; (void)out_size; (void)ws_size;
  const float* X = (const float*)d_in[0];   // [128,64,256] fp32
  const int*   y = (const int*)d_in[1];     // [128] int32

  float* s       = (float*)d_ws;            // 8192 shifts
  float* negsum  = s + N_ROWS;              // 8192 negative-exp sums
  float* negpart = negsum + N_ROWS;         // 8 x 8192 chunk partials
  float* partial = negpart + N_CHUNK * N_ROWS;  // 512 per-block partials
  float* out     = (float*)d_out;

  k_diag  <<<N_ROWS / 256, 256, 0, stream>>>(X, s);
  k_negsum<<<N_TILES,      256, 0, stream>>>(X, y, s, negpart);
  k_negred<<<N_ROWS / 256, 256, 0, stream>>>(negpart, negsum);
  k_pos   <<<N_TILES,      256, 0, stream>>>(X, y, s, negsum, partial);
  k_final <<<1,            256, 0, stream>>>(partial, out);
}